// steerable_pyramid_extractor_18193481466433
// MI455X (gfx1250) — compile-verified
//
#include <hip/hip_runtime.h>

#define B_IMG 16
#define NORI 8
#define PI_F        3.14159265358979f
#define TWO_PI_F    6.28318530717959f
#define HALF_PI_F   1.57079632679490f
#define QUART_PI_F  0.78539816339745f
// 2*sqrt(2^14 * (7!)^2 / (8 * 14!)) = 2*sqrt(1792/3003)
#define ANG_CONST   1.5449733f

typedef float v2f __attribute__((ext_vector_type(2)));
typedef float v8f __attribute__((ext_vector_type(8)));
typedef unsigned int u32x4 __attribute__((ext_vector_type(4)));
typedef int i32x4 __attribute__((ext_vector_type(4)));
typedef int i32x8 __attribute__((ext_vector_type(8)));

#if __has_builtin(__builtin_amdgcn_tensor_load_to_lds) && __has_builtin(__builtin_amdgcn_s_wait_tensorcnt)
#define HAVE_TDM 1
#endif

#ifdef HAVE_TDM
// Build a D# for a 2-D tile of f32 (data_size=4B) and issue TENSOR_LOAD_TO_LDS.
// tensor dims / strides in elements (per ISA §8: addresses scaled by data_size).
// Reads beyond tensor_dim return zero (used to zero-fill N<32 edge columns).
__device__ __forceinline__ void tdm_load_tile(unsigned ldsByteOff, const float* g,
                                              unsigned tensor_d0, unsigned tensor_d1,
                                              unsigned tile_d0, unsigned tile_d1,
                                              unsigned long long stride0)
{
    const unsigned long long ga = (unsigned long long)g;
    u32x4 g0;
    g0[0] = 1u;                                                    // D# count=1
    g0[1] = ldsByteOff;                                            // lds_addr
    g0[2] = (unsigned)(ga & 0xFFFFFFFFull);                        // global_addr lo
    g0[3] = (unsigned)((ga >> 32) & 0x01FFFFFFull) | (2u << 30);   // addr hi | type=2
    unsigned w0 = 2u << 16;                                        // data_size = 4 bytes
    unsigned w1 = (tensor_d0 & 0xFFFFu) << 16;                     // tensor_dim0[15:0] @ bits63:48
    unsigned w2 = ((tensor_d0 >> 16) & 0xFFFFu) | ((tensor_d1 & 0xFFFFu) << 16);
    unsigned w3 = ((tensor_d1 >> 16) & 0xFFFFu) | ((tile_d0 & 0xFFFFu) << 16);
    unsigned w4 = (tile_d1 & 0xFFFFu);                             // tile_dim1; tile_dim2=0
    unsigned w5 = (unsigned)(stride0 & 0xFFFFFFFFull);             // dim0 stride lo32
    unsigned w6 = (unsigned)((stride0 >> 32) & 0xFFFFull);         // dim0 stride hi16
    unsigned w7 = 0u;
    i32x8 g1;
    g1[0] = (int)w0; g1[1] = (int)w1; g1[2] = (int)w2; g1[3] = (int)w3;
    g1[4] = (int)w4; g1[5] = (int)w5; g1[6] = (int)w6; g1[7] = (int)w7;
    i32x4 z4 = {0, 0, 0, 0};
#if defined(__clang_major__) && (__clang_major__ >= 23)
    i32x8 z8 = {0, 0, 0, 0, 0, 0, 0, 0};
    __builtin_amdgcn_tensor_load_to_lds(g0, g1, z4, z4, z8, 0);
#else
    __builtin_amdgcn_tensor_load_to_lds(g0, g1, z4, z4, 0);
#endif
}
#endif

// ---------------------------------------------------------------------------
// Complex GEMM via V_WMMA_F32_16X16X4_F32 (wave32, one wave per 16x32 C slab).
// C = A * B, planar complex, row-major, square m x m operands.
// Operand panels are staged in LDS by the Tensor Data Mover (TENSORcnt), then
// fragments are read with ds_load; A fragments are reused across two N tiles.
// aImSign / bImSign = -1 conjugates that operand (for IDFT), scale at store.
// ---------------------------------------------------------------------------
#define KCMAX 64

__global__ __launch_bounds__(32)
void cgemm_wmma(const float* __restrict__ Are, const float* __restrict__ Aim, long long aStride,
                const float* __restrict__ Bre, const float* __restrict__ Bim, long long bStride,
                float* __restrict__ Cre, float* __restrict__ Cim, long long cStride,
                int M, int N, int K, float aImSign, float bImSign, float scale)
{
    (void)M;
    // Single shared block => LDS offset 0; sub-buffer offsets are constants.
    __shared__ float lds[16*KCMAX*2 + KCMAX*32*2];
    float* lAre = lds;
    float* lAim = lds + 16*KCMAX;
    float* lBre = lds + 16*KCMAX*2;
    float* lBim = lds + 16*KCMAX*2 + KCMAX*32;
#ifdef HAVE_TDM
    const unsigned offAre = 0u;
    const unsigned offAim = 16u*KCMAX*4u;
    const unsigned offBre = 16u*KCMAX*2u*4u;
    const unsigned offBim = (16u*KCMAX*2u + KCMAX*32u)*4u;
#endif

    const int lane = threadIdx.x;
    const int hf   = lane >> 4;
    const int q    = lane & 15;
    const int n0   = blockIdx.x * 32;
    const int m0   = blockIdx.y * 16;
    const long long bidx = blockIdx.z;

    const float* are = Are + bidx * aStride;
    const float* aim = Aim + bidx * aStride;
    const float* bre = Bre + bidx * bStride;
    const float* bim = Bim + bidx * bStride;

    const bool two = (n0 + 16) < N;           // second N tile present (uniform)
    const int kc = (K < KCMAX) ? K : KCMAX;

    v8f aR0 = {0.f,0.f,0.f,0.f,0.f,0.f,0.f,0.f};
    v8f aI0 = {0.f,0.f,0.f,0.f,0.f,0.f,0.f,0.f};
    v8f aR1 = {0.f,0.f,0.f,0.f,0.f,0.f,0.f,0.f};
    v8f aI1 = {0.f,0.f,0.f,0.f,0.f,0.f,0.f,0.f};

    for (int k0 = 0; k0 < K; k0 += kc) {
#ifdef HAVE_TDM
        // A panel: 16 rows x kc cols (row stride K); B panel: kc rows x 32 cols
        // (row stride N, cols beyond N zero-filled by TDM OOB rule).
        tdm_load_tile(offAre, are + (long long)m0 * K + k0, (unsigned)(K - k0), 16u,
                      (unsigned)kc, 16u, (unsigned long long)K);
        tdm_load_tile(offAim, aim + (long long)m0 * K + k0, (unsigned)(K - k0), 16u,
                      (unsigned)kc, 16u, (unsigned long long)K);
        tdm_load_tile(offBre, bre + (long long)k0 * N + n0, (unsigned)(N - n0), (unsigned)(K - k0),
                      32u, (unsigned)kc, (unsigned long long)N);
        tdm_load_tile(offBim, bim + (long long)k0 * N + n0, (unsigned)(N - n0), (unsigned)(K - k0),
                      32u, (unsigned)kc, (unsigned long long)N);
        __builtin_amdgcn_s_wait_tensorcnt(0);
#else
        for (int idx = lane; idx < 16 * kc; idx += 32) {
            const int r = idx / kc, c = idx % kc;
            lAre[idx] = are[(long long)(m0 + r) * K + k0 + c];
            lAim[idx] = aim[(long long)(m0 + r) * K + k0 + c];
        }
        for (int idx = lane; idx < kc * 32; idx += 32) {
            const int r = idx / 32, c = idx % 32;
            float vr = 0.f, vi = 0.f;
            if (n0 + c < N) {
                vr = bre[(long long)(k0 + r) * N + n0 + c];
                vi = bim[(long long)(k0 + r) * N + n0 + c];
            }
            lBre[idx] = vr; lBim[idx] = vi;
        }
#endif
        __syncthreads();

        for (int k = 0; k < kc; k += 4) {
            const int ka = k + 2 * hf;
            // A fragment: lane holds (row=m0+q, K=ka..ka+1) -> ds_load_b64
            v2f ar = *(const v2f*)&lAre[q * kc + ka];
            v2f ai = *(const v2f*)&lAim[q * kc + ka];
            ai = ai * aImSign;
            // B fragment tile0: (K=ka..ka+1, col=q)
            v2f br0, bi0;
            br0.x = lBre[ka * 32 + q];       br0.y = lBre[(ka + 1) * 32 + q];
            bi0.x = lBim[ka * 32 + q];       bi0.y = lBim[(ka + 1) * 32 + q];
            bi0 = bi0 * bImSign;
            v2f nbi0 = -bi0;   // f32 WMMA NEG only encodes C-negate -> negate in VALU
            aR0 = __builtin_amdgcn_wmma_f32_16x16x4_f32(false, ar, false, br0,  (short)0, aR0, false, false);
            aR0 = __builtin_amdgcn_wmma_f32_16x16x4_f32(false, ai, false, nbi0, (short)0, aR0, false, false);
            aI0 = __builtin_amdgcn_wmma_f32_16x16x4_f32(false, ar, false, bi0,  (short)0, aI0, false, false);
            aI0 = __builtin_amdgcn_wmma_f32_16x16x4_f32(false, ai, false, br0,  (short)0, aI0, false, false);
            if (two) {
                v2f br1, bi1;
                br1.x = lBre[ka * 32 + 16 + q]; br1.y = lBre[(ka + 1) * 32 + 16 + q];
                bi1.x = lBim[ka * 32 + 16 + q]; bi1.y = lBim[(ka + 1) * 32 + 16 + q];
                bi1 = bi1 * bImSign;
                v2f nbi1 = -bi1;
                aR1 = __builtin_amdgcn_wmma_f32_16x16x4_f32(false, ar, false, br1,  (short)0, aR1, false, false);
                aR1 = __builtin_amdgcn_wmma_f32_16x16x4_f32(false, ai, false, nbi1, (short)0, aR1, false, false);
                aI1 = __builtin_amdgcn_wmma_f32_16x16x4_f32(false, ar, false, bi1,  (short)0, aI1, false, false);
                aI1 = __builtin_amdgcn_wmma_f32_16x16x4_f32(false, ai, false, br1,  (short)0, aI1, false, false);
            }
        }
        __syncthreads();
    }

    float* cre = Cre + bidx * cStride;
    float* cim = Cim + bidx * cStride;
#pragma unroll
    for (int v = 0; v < 8; ++v) {
        const int row = m0 + v + 8 * hf;
        cre[(long long)row * N + n0 + q] = scale * aR0[v];
        cim[(long long)row * N + n0 + q] = scale * aI0[v];
        if (two) {
            cre[(long long)row * N + n0 + 16 + q] = scale * aR1[v];
            cim[(long long)row * N + n0 + 16 + q] = scale * aI1[v];
        }
    }
}

// ---------------------------------------------------------------------------
// Mask math (pyrtools SteerablePyramidFreq, evaluated at unshifted frequencies;
// after central crops the normalized coordinate is always f/128 for n=256).
// ---------------------------------------------------------------------------
__device__ __forceinline__ float rc_fn(float logr, float pos) {
    float X = (logr - pos) * HALF_PI_F - QUART_PI_F;
    X = fminf(fmaxf(X, -HALF_PI_F), 0.0f);
    float c = __cosf(X);
    return c * c;
}
__device__ __forceinline__ float hi_mask(float logr, float pos) {
    return sqrtf(fmaxf(rc_fn(logr, pos), 0.0f));
}
__device__ __forceinline__ float lo_mask(float logr, float pos) {
    return sqrtf(fmaxf(1.0f - rc_fn(logr, pos), 0.0f));
}
__device__ __forceinline__ float cum_to(float logr, int levels) {
    float c = lo_mask(logr, -0.5f);
    for (int j = 0; j < levels; ++j) c *= lo_mask(logr, -1.5f - (float)j);
    return c;
}
__device__ __forceinline__ float logr_at(int fr, int fc) {
    float r2 = (float)(fr*fr + fc*fc);
    float rad = (r2 > 0.0f) ? (sqrtf(r2) * (1.0f/128.0f)) : (1.0f/128.0f); // DC fix
    return log2f(rad);
}

// ---------------------------------------------------------------------------
// Pointwise kernels
// ---------------------------------------------------------------------------
__global__ void gen_dft_kernel(float* __restrict__ fre, float* __restrict__ fim, int n) {
    long long i = (long long)blockIdx.x * 256 + threadIdx.x;
    long long nn = (long long)n * n;
    if (i >= nn) return;
    int j = (int)(i / n), k = (int)(i % n);
    int t = (int)(((long long)j * k) % n);
    float ang = -TWO_PI_F * (float)t / (float)n;
    float sv, cv; __sincosf(ang, &sv, &cv);
    fre[i] = cv; fim[i] = sv;
}

__global__ void pack_r2c_kernel(const float* __restrict__ src,
                                float* __restrict__ re, float* __restrict__ im, long long total) {
    long long i = (long long)blockIdx.x * 256 + threadIdx.x;
    if (i >= total) return;
    re[i] = src[i]; im[i] = 0.0f;
}

// kind: 0=hi0  2=cumhi_l^2  3=cum_low  4=cum_low^2
__global__ void radial_mask_kernel(const float* __restrict__ dre, const float* __restrict__ dim_,
                                   float* __restrict__ ore, float* __restrict__ oim,
                                   int m, int kind, int level, long long total) {
    long long i = (long long)blockIdx.x * 256 + threadIdx.x;
    if (i >= total) return;
    const long long mm = (long long)m * m;
    const int p = (int)(i % mm);
    const int r = p / m, c = p % m;
    const int fr = (r < (m >> 1)) ? r : r - m;
    const int fc = (c < (m >> 1)) ? c : c - m;
    const float lr = logr_at(fr, fc);
    float mask;
    if (kind == 0)      { mask = hi_mask(lr, -0.5f); }
    else if (kind == 2) { float ch = cum_to(lr, level) * hi_mask(lr, -1.5f - (float)level); mask = ch * ch; }
    else if (kind == 3) { mask = cum_to(lr, 4); }
    else                { float cl = cum_to(lr, 4); mask = cl * cl; }
    ore[i] = dre[i] * mask;
    oim[i] = dim_[i] * mask;
}

// banddft = i * D * (cumhi_l * A_ori)   (phase_const (-1j)^7 == +1j)
__global__ void band_mask_kernel(const float* __restrict__ dre, const float* __restrict__ dim_,
                                 float* __restrict__ ore, float* __restrict__ oim,
                                 int m, int level, long long total) {
    long long i = (long long)blockIdx.x * 256 + threadIdx.x;
    if (i >= total) return;
    const long long mm = (long long)m * m;
    const long long imgi = i / mm;
    const int ori = (int)(imgi & 7);
    const int p = (int)(i % mm);
    const int r = p / m, c = p % m;
    const int fr = (r < (m >> 1)) ? r : r - m;
    const int fc = (c < (m >> 1)) ? c : c - m;
    const float lr = logr_at(fr, fc);
    const float cumhi = cum_to(lr, level) * hi_mask(lr, -1.5f - (float)level);
    const float ang = atan2f((float)fr, (float)fc);   // atan2(Y, X)
    float t = PI_F + ang - PI_F * (float)ori / 8.0f;
    t = t - TWO_PI_F * floorf(t / TWO_PI_F);          // np.mod into [0, 2pi)
    const float a = t - PI_F;
    float A = 0.0f;
    if (fabsf(a) < HALF_PI_F) {
        const float ca = __cosf(a);
        const float ca2 = ca * ca;
        A = ANG_CONST * ca2 * ca2 * ca2 * ca;          // cos^7
    }
    const float Mk = cumhi * A;
    const long long src = (imgi >> 3) * mm + p;
    const float xr = dre[src], xi = dim_[src];
    ore[i] = -xi * Mk;    // (a+bi)*i = -b + ai
    oim[i] =  xr * Mk;
}

// shifted central-crop == corner gather in unshifted layout
__global__ void crop_spec_kernel(const float* __restrict__ ir, const float* __restrict__ ii,
                                 float* __restrict__ or_, float* __restrict__ oi,
                                 int m, long long total) {
    long long i = (long long)blockIdx.x * 256 + threadIdx.x;
    if (i >= total) return;
    const int m2 = m >> 1;
    const long long mm2 = (long long)m2 * m2;
    const long long img = i / mm2;
    const int p = (int)(i % mm2);
    const int r = p / m2, c = p % m2;
    const int fr = (r < (m2 >> 1)) ? r : r - m2;
    const int fc = (c < (m2 >> 1)) ? c : c - m2;
    const int sr = (fr < 0) ? fr + m : fr;
    const int sc = (fc < 0) ? fc + m : fc;
    const long long s = img * (long long)m * m + (long long)sr * m + sc;
    or_[i] = ir[s]; oi[i] = ii[s];
}

// zero-pad spectrum m -> 2m (expand2), self-initializing
__global__ void upsample_spec_kernel(const float* __restrict__ ir, const float* __restrict__ ii,
                                     float* __restrict__ or_, float* __restrict__ oi,
                                     int m, long long total) {
    long long i = (long long)blockIdx.x * 256 + threadIdx.x;
    if (i >= total) return;
    const int M = m << 1;
    const long long MM = (long long)M * M;
    const long long img = i / MM;
    const int p = (int)(i % MM);
    const int r = p / M, c = p % M;
    const int fr = (r < m) ? r : r - M;
    const int fc = (c < m) ? c : c - M;
    float vr = 0.0f, vi = 0.0f;
    const int h = m >> 1;
    if (fr >= -h && fr < h && fc >= -h && fc < h) {
        const int sr = (fr < 0) ? fr + m : fr;
        const int sc = (fc < 0) ? fc + m : fc;
        const long long s = img * (long long)m * m + (long long)sr * m + sc;
        vr = ir[s]; vi = ii[s];
    }
    or_[i] = vr; oi[i] = vi;
}

__global__ void dphase_store_kernel(const float* __restrict__ ir, const float* __restrict__ ii,
                                    float* __restrict__ out, long long total) {
    long long i = (long long)blockIdx.x * 256 + threadIdx.x;
    if (i >= total) return;
    const float re = ir[i], im = ii[i];
    const float rr = sqrtf(re * re + im * im);
    float outr = 0.0f, outi = 0.0f;
    if (rr > 0.0f) { outr = (re * re - im * im) / rr; outi = (2.0f * re * im) / rr; }
    out[2 * i] = outr; out[2 * i + 1] = outi;
}

__global__ void store_real_kernel(const float* __restrict__ re, float* __restrict__ out, long long total) {
    long long i = (long long)blockIdx.x * 256 + threadIdx.x;
    if (i >= total) return;
    out[i] = re[i];
}
__global__ void store_cplx_kernel(const float* __restrict__ re, const float* __restrict__ im,
                                  float* __restrict__ out, long long total) {
    long long i = (long long)blockIdx.x * 256 + threadIdx.x;
    if (i >= total) return;
    out[2 * i] = re[i]; out[2 * i + 1] = im[i];
}
__global__ void store_real_as_cplx_kernel(const float* __restrict__ re, float* __restrict__ out, long long total) {
    long long i = (long long)blockIdx.x * 256 + threadIdx.x;
    if (i >= total) return;
    out[2 * i] = re[i]; out[2 * i + 1] = 0.0f;
}
__global__ void add_store_real_kernel(const float* __restrict__ a, const float* __restrict__ b,
                                      float* __restrict__ lpbuf, float* __restrict__ out, long long total) {
    long long i = (long long)blockIdx.x * 256 + threadIdx.x;
    if (i >= total) return;
    const float v = a[i] + b[i];
    lpbuf[i] = v; out[i] = v;
}

// ---------------------------------------------------------------------------
// Host-side 2D (I)FFT as two WMMA GEMM passes: Y = F * (X * F), F symmetric.
// pass1: x -> t, pass2: t -> y  (x/y may alias, t must differ)
// ---------------------------------------------------------------------------
static void fft2(hipStream_t s, int m, int nb,
                 const float* xr, const float* xi,
                 float* tr, float* ti,
                 float* yr, float* yi,
                 const float* Fr, const float* Fi, bool inverse)
{
    const float sgn = inverse ? -1.0f : 1.0f;
    const float sc  = inverse ? (1.0f / (float)m) : 1.0f;
    dim3 g((m + 31) / 32, m / 16, nb), blk(32);
    const long long mm = (long long)m * m;
    cgemm_wmma<<<g, blk, 0, s>>>(xr, xi, mm, Fr, Fi, 0LL, tr, ti, mm, m, m, m, 1.0f, sgn, sc);
    cgemm_wmma<<<g, blk, 0, s>>>(Fr, Fi, 0LL, tr, ti, mm, yr, yi, mm, m, m, m, sgn, 1.0f, sc);
}

extern "C" void kernel_launch(void* const* d_in, const int* in_sizes, int n_in,
                              void* d_out, int out_size, void* d_ws, size_t ws_size,
                              hipStream_t stream)
{
    (void)in_sizes; (void)n_in; (void)out_size; (void)ws_size;
    const float* img = (const float*)d_in[0];
    float* out = (float*)d_out;
    float* W = (float*)d_ws;

    const int ms[5] = {256, 128, 64, 32, 16};

    // ---- workspace layout ----
    size_t off = 0;
    auto alloc = [&](size_t n) -> float* { float* p = W + off; off += (n + 63) & ~(size_t)63; return p; };

    float *Fre[5], *Fim[5];
    for (int i = 0; i < 5; ++i) { size_t nn = (size_t)ms[i] * ms[i]; Fre[i] = alloc(nn); Fim[i] = alloc(nn); }
    float *Dre[5], *Dim[5];
    for (int l = 0; l < 5; ++l) { size_t nn = (size_t)B_IMG * ms[l] * ms[l]; Dre[l] = alloc(nn); Dim[l] = alloc(nn); }
    const size_t TMP = (size_t)B_IMG * NORI * 256 * 256;
    float* t1r = alloc(TMP); float* t1i = alloc(TMP);
    float* t2r = alloc(TMP); float* t2i = alloc(TMP);
    float *BdR[4] = {0,0,0,0}, *BdI[4] = {0,0,0,0};
    for (int l = 1; l < 4; ++l) { size_t nn = (size_t)B_IMG * NORI * ms[l] * ms[l]; BdR[l] = alloc(nn); BdI[l] = alloc(nn); }
    float* Brec[4];
    for (int l = 0; l < 4; ++l) { Brec[l] = alloc((size_t)B_IMG * ms[l] * ms[l]); }
    float* rlow  = alloc((size_t)B_IMG * 256);
    float* rclow = alloc((size_t)B_IMG * 256);
    float* lp    = alloc((size_t)B_IMG * 256 * 256);

    // ---- output offsets (reference return order) ----
    size_t o = 0;
    auto take = [&](size_t n) { size_t r = o; o += n; return r; };
    size_t o_fc[4];
    o_fc[0] = take((size_t)B_IMG * NORI * 32  * 32  * 2);
    o_fc[1] = take((size_t)B_IMG * NORI * 64  * 64  * 2);
    o_fc[2] = take((size_t)B_IMG * NORI * 128 * 128 * 2);
    o_fc[3] = take((size_t)B_IMG * NORI * 256 * 256 * 2);
    size_t o_rlow  = take((size_t)B_IMG * 16 * 16);
    size_t o_rhigh = take((size_t)B_IMG * 256 * 256);
    size_t o_lp[5];
    o_lp[0] = take((size_t)B_IMG * 16  * 16);
    o_lp[1] = take((size_t)B_IMG * 32  * 32);
    o_lp[2] = take((size_t)B_IMG * 64  * 64);
    o_lp[3] = take((size_t)B_IMG * 128 * 128);
    o_lp[4] = take((size_t)B_IMG * 256 * 256);
    size_t o_co[4];
    o_co[0] = take((size_t)B_IMG * 32 * 32 * 2);
    o_co[1] = take((size_t)B_IMG * NORI * 64  * 64  * 2);
    o_co[2] = take((size_t)B_IMG * NORI * 128 * 128 * 2);
    o_co[3] = take((size_t)B_IMG * NORI * 256 * 256 * 2);

    auto g1 = [](long long t) { return dim3((unsigned)((t + 255) / 256)); };
    dim3 b256(256);

    // 0) DFT matrix tables
    for (int i = 0; i < 5; ++i) {
        long long nn = (long long)ms[i] * ms[i];
        gen_dft_kernel<<<g1(nn), b256, 0, stream>>>(Fre[i], Fim[i], ms[i]);
    }

    // 1) dft = fftshift(fft2(image))  (kept unshifted; masks use f-coords)
    {
        long long tot = (long long)B_IMG * 256 * 256;
        pack_r2c_kernel<<<g1(tot), b256, 0, stream>>>(img, t1r, t1i, tot);
        fft2(stream, 256, B_IMG, t1r, t1i, t2r, t2i, Dre[0], Dim[0], Fre[0], Fim[0], false);
    }

    // 2) resid_high = real(ifft2(dft * hi0))
    {
        long long tot = (long long)B_IMG * 256 * 256;
        radial_mask_kernel<<<g1(tot), b256, 0, stream>>>(Dre[0], Dim[0], t1r, t1i, 256, 0, 0, tot);
        fft2(stream, 256, B_IMG, t1r, t1i, t2r, t2i, t1r, t1i, Fre[0], Fim[0], true);
        store_real_kernel<<<g1(tot), b256, 0, stream>>>(t1r, out + o_rhigh, tot);
    }

    // 3) pyramid levels
    for (int l = 0; l < 4; ++l) {
        const int m = ms[l];
        const long long mm = (long long)m * m;
        // bands: i * imdft * cumhi * A_ori, then ifft2 (B*8 images)
        long long tb = (long long)B_IMG * NORI * mm;
        band_mask_kernel<<<g1(tb), b256, 0, stream>>>(Dre[l], Dim[l], t1r, t1i, m, l, tb);
        fft2(stream, m, B_IMG * NORI, t1r, t1i, t2r, t2i, t1r, t1i, Fre[l], Fim[l], true);
        store_cplx_kernel<<<g1(tb), b256, 0, stream>>>(t1r, t1i, out + o_fc[3 - l], tb);
        if (l >= 1) {   // keep planar copies for expand2 later
            hipMemcpyAsync(BdR[l], t1r, (size_t)tb * sizeof(float), hipMemcpyDeviceToDevice, stream);
            hipMemcpyAsync(BdI[l], t1i, (size_t)tb * sizeof(float), hipMemcpyDeviceToDevice, stream);
        }
        // band_recon = real(ifft2(imdft * cumhi^2))
        long long trc = (long long)B_IMG * mm;
        radial_mask_kernel<<<g1(trc), b256, 0, stream>>>(Dre[l], Dim[l], t1r, t1i, m, 2, l, trc);
        fft2(stream, m, B_IMG, t1r, t1i, t2r, t2i, t1r, t1i, Fre[l], Fim[l], true);
        hipMemcpyAsync(Brec[l], t1r, (size_t)trc * sizeof(float), hipMemcpyDeviceToDevice, stream);
        // crop spectrum to next level
        long long tc = (long long)B_IMG * (m / 2) * (m / 2);
        crop_spec_kernel<<<g1(tc), b256, 0, stream>>>(Dre[l], Dim[l], Dre[l + 1], Dim[l + 1], m, tc);
    }

    // 4) resid_low / recon_low at 16x16
    {
        long long tot = (long long)B_IMG * 256;
        radial_mask_kernel<<<g1(tot), b256, 0, stream>>>(Dre[4], Dim[4], t1r, t1i, 16, 3, 0, tot);
        fft2(stream, 16, B_IMG, t1r, t1i, t2r, t2i, t1r, t1i, Fre[4], Fim[4], true);
        hipMemcpyAsync(rlow, t1r, (size_t)tot * sizeof(float), hipMemcpyDeviceToDevice, stream);
        store_real_kernel<<<g1(tot), b256, 0, stream>>>(t1r, out + o_rlow, tot);

        radial_mask_kernel<<<g1(tot), b256, 0, stream>>>(Dre[4], Dim[4], t1r, t1i, 16, 4, 0, tot);
        fft2(stream, 16, B_IMG, t1r, t1i, t2r, t2i, t1r, t1i, Fre[4], Fim[4], true);
        hipMemcpyAsync(rclow, t1r, (size_t)tot * sizeof(float), hipMemcpyDeviceToDevice, stream);
        store_real_kernel<<<g1(tot), b256, 0, stream>>>(t1r, out + o_lp[0], tot);
    }

    // 5) coarser[0] = real(expand2(resid_low)) as complex
    {
        long long tin = (long long)B_IMG * 256;
        pack_r2c_kernel<<<g1(tin), b256, 0, stream>>>(rlow, t1r, t1i, tin);
        fft2(stream, 16, B_IMG, t1r, t1i, t2r, t2i, t1r, t1i, Fre[4], Fim[4], false);
        long long tout = (long long)B_IMG * 32 * 32;
        upsample_spec_kernel<<<g1(tout), b256, 0, stream>>>(t1r, t1i, t2r, t2i, 16, tout);
        fft2(stream, 32, B_IMG, t2r, t2i, t1r, t1i, t2r, t2i, Fre[3], Fim[3], true);
        store_real_as_cplx_kernel<<<g1(tout), b256, 0, stream>>>(t2r, out + o_co[0], tout);
    }

    // 6) coarser[sf+1] = double_phase(expand2(bands[3-sf])), sf = 0..2
    for (int sf = 0; sf < 3; ++sf) {
        const int l = 3 - sf;
        const int m = ms[l];
        fft2(stream, m, B_IMG * NORI, BdR[l], BdI[l], t1r, t1i, t2r, t2i, Fre[l], Fim[l], false);
        long long tout = (long long)B_IMG * NORI * (2 * m) * (2 * m);
        upsample_spec_kernel<<<g1(tout), b256, 0, stream>>>(t2r, t2i, t1r, t1i, m, tout);
        fft2(stream, 2 * m, B_IMG * NORI, t1r, t1i, t2r, t2i, t1r, t1i, Fre[l - 1], Fim[l - 1], true);
        dphase_store_kernel<<<g1(tout), b256, 0, stream>>>(t1r, t1i, out + o_co[sf + 1], tout);
    }

    // 7) lowpass reconstruction chain
    hipMemcpyAsync(lp, rclow, (size_t)B_IMG * 256 * sizeof(float), hipMemcpyDeviceToDevice, stream);
    {
        int m = 16;
        for (int sf = 0; sf < 4; ++sf) {
            const int li  = 4 - sf;   // table index for m
            const int li2 = 3 - sf;   // table index for 2m
            long long tin = (long long)B_IMG * m * m;
            pack_r2c_kernel<<<g1(tin), b256, 0, stream>>>(lp, t1r, t1i, tin);
            fft2(stream, m, B_IMG, t1r, t1i, t2r, t2i, t1r, t1i, Fre[li], Fim[li], false);
            long long tout = (long long)B_IMG * (2 * m) * (2 * m);
            upsample_spec_kernel<<<g1(tout), b256, 0, stream>>>(t1r, t1i, t2r, t2i, m, tout);
            fft2(stream, 2 * m, B_IMG, t2r, t2i, t1r, t1i, t2r, t2i, Fre[li2], Fim[li2], true);
            add_store_real_kernel<<<g1(tout), b256, 0, stream>>>(t2r, Brec[3 - sf], lp, out + o_lp[sf + 1], tout);
            m *= 2;
        }
    }
}